// VectorQuantizer_24008867184954
// MI455X (gfx1250) — compile-verified
//
#include <hip/hip_runtime.h>
#include <hip/hip_bf16.h>

// VQ-VAE vector quantizer for MI455X (gfx1250, wave32).
// Dominant cost: 32768x1024x256 fp32 GEMM (-2 x.e) -> compute-bound on WMMA.
// Uses V_WMMA_F32_16X16X4_F32 to match the fp32 reference argmin exactly.

typedef __attribute__((ext_vector_type(2))) float v2f;
typedef __attribute__((ext_vector_type(8))) float v8f;

#define DIM   256      // embedding dim (GEMM K-dimension)
#define KC    1024     // number of codes (GEMM N-dimension)
#define HWSZ  1024     // H*W = 32*32
#define NROWS 32768    // B*H*W
#define NELEM 8388608  // NROWS*DIM

// ---------------------------------------------------------------------------
// Kernel 1: per-code squared norms  ||e_k||^2   (grid=1024, block=64)
// ---------------------------------------------------------------------------
__global__ void vq_norms_kernel(const float* __restrict__ codebook,
                                float* __restrict__ norms) {
  const int k = blockIdx.x;
  const float4 v = *(const float4*)(codebook + k * DIM + threadIdx.x * 4);
  float s = v.x * v.x + v.y * v.y + v.z * v.z + v.w * v.w;
#pragma unroll
  for (int off = 16; off >= 1; off >>= 1) s += __shfl_xor(s, off, 32);
  __shared__ float tsum[2];
  if ((threadIdx.x & 31) == 0) tsum[threadIdx.x >> 5] = s;
  __syncthreads();
  if (threadIdx.x == 0) norms[k] = tsum[0] + tsum[1];
}

// ---------------------------------------------------------------------------
// Kernel 2: WMMA GEMM + fused argmin.
// grid = NROWS/128 = 256 blocks, 256 threads = 8 waves.
// Each wave owns a 16-row tile; A panel (16x256 fp32) held in 128 VGPRs/lane.
// Block stages each 16-code x 256 B panel into LDS (row pad 260 floats ->
// conflict-free ds_load_b64 fragment reads across all 64 banks).
// ---------------------------------------------------------------------------
__global__ __launch_bounds__(256, 1) void vq_argmin_kernel(
    const float* __restrict__ latent, const float* __restrict__ codebook,
    const float* __restrict__ norms, int* __restrict__ idxg) {
  __shared__ float ldsB[16 * 260];

  const int tid  = threadIdx.x;
  const int wave = tid >> 5;
  const int lane = tid & 31;
  const int lrow = lane & 15;   // M (for A) / N (for B) within the 16-tile
  const int hsel = lane >> 4;   // K-half select per ISA lane layout

  const int rowbase = blockIdx.x * 128 + wave * 16;
  const int row = rowbase + lrow;        // never crosses a batch boundary
  const int b   = row >> 10;
  const int inb = row & 1023;
  // flat[row][d] = latent[b*DIM*HWSZ + d*HWSZ + inb]; 16 consecutive lanes ->
  // 16 consecutive floats within a d-plane (coalesced).
  const float* xbase = latent + b * (DIM * HWSZ) + inb;

  // A fragments in registers: a[s] covers K = 4s..4s+3 split across lane halves
  // (V0: K=4s (lanes 0-15) / 4s+2 (lanes 16-31); V1: 4s+1 / 4s+3).
  v2f a[64];
#pragma unroll
  for (int s = 0; s < 64; ++s) {
    const int kk = 4 * s + 2 * hsel;
    v2f av;
    av.x = xbase[kk * HWSZ];
    av.y = xbase[(kk + 1) * HWSZ];
    a[s] = av;
  }

  float minv[8];
  int   mini[8];
#pragma unroll
  for (int j = 0; j < 8; ++j) { minv[j] = 3.0e38f; mini[j] = 0; }

  for (int t = 0; t < 64; ++t) {
    const int c0 = t * 16;
    __syncthreads();
    // Cooperative stage of 16 codebook rows (16 KB) via b128 loads.
#pragma unroll
    for (int i = 0; i < 4; ++i) {
      const int fidx = i * 256 + tid;          // float4 index 0..1023
      const int r  = fidx >> 6;                // code row 0..15
      const int j4 = fidx & 63;                // float4 within row
      const float4 vv =
          *(const float4*)(codebook + (c0 + r) * DIM + 4 * j4);
      *(float4*)(&ldsB[r * 260 + 4 * j4]) = vv;
    }
    __syncthreads();

    v8f c = {0.f, 0.f, 0.f, 0.f, 0.f, 0.f, 0.f, 0.f};
    const float* bbase = &ldsB[lrow * 260 + 2 * hsel];
#pragma unroll
    for (int s = 0; s < 64; ++s) {
      const v2f bv = *(const v2f*)(bbase + 4 * s);
      c = __builtin_amdgcn_wmma_f32_16x16x4_f32(
          /*neg_a=*/false, a[s], /*neg_b=*/false, bv,
          /*c_mod=*/(short)0, c, /*reuse_a=*/false, /*reuse_b=*/false);
    }

    // C layout: lane n holds column c0+(n&15); VGPR j -> row j (+8 for hi half).
    const int col  = c0 + lrow;
    const float nr = norms[col];
#pragma unroll
    for (int j = 0; j < 8; ++j) {
      const float dist = fmaf(-2.0f, c[j], nr);   // ||e||^2 - 2 x.e
      if (dist < minv[j]) { minv[j] = dist; mini[j] = col; }
    }
  }

  // Reduce (min, idx) across the 16 lanes of each half (wave32 shuffles);
  // tie-break on smaller index to match jnp.argmin.
#pragma unroll
  for (int j = 0; j < 8; ++j) {
    float v = minv[j];
    int  ix = mini[j];
#pragma unroll
    for (int off = 8; off >= 1; off >>= 1) {
      const float ov = __shfl_xor(v, off, 32);
      const int   oi = __shfl_xor(ix, off, 32);
      if (ov < v || (ov == v && oi < ix)) { v = ov; ix = oi; }
    }
    if (lrow == 0) idxg[rowbase + hsel * 8 + j] = ix;
  }
}

// ---------------------------------------------------------------------------
// Kernel 3: gather codebook[idx] into [B,D,H,W] output + accumulate (q-x)^2.
// grid = NELEM/256 blocks. Codebook (1 MB) lives in L2; gathers are cheap.
// ---------------------------------------------------------------------------
__global__ void vq_gather_loss_kernel(const float* __restrict__ latent,
                                      const float* __restrict__ codebook,
                                      const int* __restrict__ idxg,
                                      float* __restrict__ out,
                                      float* __restrict__ acc) {
  const int tid = blockIdx.x * 256 + threadIdx.x;  // linear over [B,D,H,W]
  const int inb = tid & 1023;
  const int d   = (tid >> 10) & 255;
  const int b   = tid >> 18;
  const int n   = (b << 10) | inb;
  const int code = idxg[n];
  const float q = codebook[code * DIM + d];
  const float x = latent[tid];
  out[tid] = q;                      // q_st == q numerically
  const float diff = q - x;
  float s = diff * diff;
#pragma unroll
  for (int off = 16; off >= 1; off >>= 1) s += __shfl_xor(s, off, 32);
  __shared__ float wsum[8];
  if ((threadIdx.x & 31) == 0) wsum[threadIdx.x >> 5] = s;
  __syncthreads();
  if (threadIdx.x == 0) {
    float t = 0.f;
#pragma unroll
    for (int i = 0; i < 8; ++i) t += wsum[i];
    atomicAdd(acc, t);
  }
}

// loss = embedding + 0.25*commitment = 1.25 * mean((q-x)^2)
__global__ void vq_finalize_kernel(const float* __restrict__ acc,
                                   float* __restrict__ loss_out) {
  *loss_out = 1.25f * (*acc) / (float)NELEM;
}

// ---------------------------------------------------------------------------
extern "C" void kernel_launch(void* const* d_in, const int* in_sizes, int n_in,
                              void* d_out, int out_size, void* d_ws,
                              size_t ws_size, hipStream_t stream) {
  const float* latent   = (const float*)d_in[0];   // [32,256,32,32]
  const float* codebook = (const float*)d_in[1];   // [1024,256]
  float* out  = (float*)d_out;                     // NELEM q values + 1 loss
  float* f_ws = (float*)d_ws;
  float* norms = f_ws;                       // 1024 floats
  float* acc   = f_ws + 1024;                // 1 float
  int*   idxg  = (int*)(f_ws + 1088);        // 32768 ints (64B aligned)

  hipMemsetAsync(acc, 0, sizeof(float), stream);
  vq_norms_kernel<<<KC, 64, 0, stream>>>(codebook, norms);
  vq_argmin_kernel<<<NROWS / 128, 256, 0, stream>>>(latent, codebook, norms,
                                                    idxg);
  vq_gather_loss_kernel<<<NELEM / 256, 256, 0, stream>>>(latent, codebook,
                                                         idxg, out, acc);
  vq_finalize_kernel<<<1, 1, 0, stream>>>(acc, out + (out_size - 1));
}